// Rankloss_11261404250344
// MI455X (gfx1250) — compile-verified
//
#include <hip/hip_runtime.h>
#include <stdint.h>

// Rank loss over [B, M=4096] fp32. Bandwidth-bound: 256 MB read once (~11 us
// floor at 23.3 TB/s). Strategy: one workgroup per row, stage the 16 KB row
// into LDS with CDNA5 async global->LDS copies (ASYNCcnt path, non-temporal
// hint since the 256 MB stream exceeds the 192 MB L2 and is read exactly
// once), then compute softmax max / sum-exp / hinge loss entirely from
// LDS + registers. Deterministic two-stage final reduction (no float atomics).

#define MARGIN   0.1f
#define M_DIM    4096
#define BLK      256
#define NWAVE    (BLK / 32)   // 8 wave32s per block

// LDS byte offset of a pointer into a __shared__ object (addrspace(3) value).
__device__ __forceinline__ unsigned lds_addr_of(const void* p) {
    return (unsigned)(uintptr_t)(const __attribute__((address_space(3))) void*)p;
}

__device__ __forceinline__ float wave_max(float v) {
#pragma unroll
    for (int o = 16; o > 0; o >>= 1)
        v = fmaxf(v, __shfl_xor(v, o, 32));
    return v;
}

__device__ __forceinline__ float wave_sum(float v) {
#pragma unroll
    for (int o = 16; o > 0; o >>= 1)
        v += __shfl_xor(v, o, 32);
    return v;
}

__global__ __launch_bounds__(BLK) void rankloss_rows(
    const float* __restrict__ X,        // [B, 4096]
    const int*   __restrict__ gt,       // [B] ground-truth column
    float*       __restrict__ row_loss) // [B] per-row hinge sums
{
    __shared__ float srow[M_DIM];    // 16 KB staged row
    __shared__ float sred[NWAVE];
    __shared__ float s_max, s_sum;

    const int tid  = threadIdx.x;
    const int row  = blockIdx.x;
    const int lane = tid & 31;
    const int wav  = tid >> 5;

    // ---- stage row global -> LDS via async copies (CDNA5 ASYNCcnt path) ----
    // Non-temporal: single-use stream larger than L2, don't cache it.
    {
        const unsigned           lbase = lds_addr_of(srow);
        const unsigned long long gbase =
            (unsigned long long)(X + (size_t)row * M_DIM);
#pragma unroll
        for (int i = 0; i < 4; ++i) {
            // iteration i: block covers elements [i*1024, i*1024+1024),
            // thread t moves 16 bytes at element i*1024 + t*4 (coalesced b128)
            const unsigned           eoff = (unsigned)(i * 1024 + tid * 4) * 4u;
            const unsigned           loff = lbase + eoff;
            const unsigned long long ga   = gbase + eoff;
            asm volatile("global_load_async_to_lds_b128 %0, %1, off th:TH_LOAD_NT"
                         :: "v"(loff), "v"(ga)
                         : "memory");
        }
        asm volatile("s_wait_asynccnt 0" ::: "memory");
    }
    __syncthreads();  // all 8 waves' async copies have landed in LDS

    // ---- pull 16 elements/thread into registers, tracking max ----
    float r[16];
    float m = -3.402823466e+38f;
#pragma unroll
    for (int i = 0; i < 4; ++i) {
        const float4 v = *reinterpret_cast<const float4*>(&srow[i * 1024 + tid * 4]);
        r[i * 4 + 0] = v.x; r[i * 4 + 1] = v.y;
        r[i * 4 + 2] = v.z; r[i * 4 + 3] = v.w;
        m = fmaxf(m, fmaxf(fmaxf(v.x, v.y), fmaxf(v.z, v.w)));
    }

    // ---- block-reduce max ----
    m = wave_max(m);
    if (lane == 0) sred[wav] = m;
    __syncthreads();
    if (tid == 0) {
        float t = sred[0];
#pragma unroll
        for (int w = 1; w < NWAVE; ++w) t = fmaxf(t, sred[w]);
        s_max = t;
    }
    __syncthreads();
    const float rmax = s_max;

    // ---- exp and block-reduce sum (keep exp values in registers) ----
    float s = 0.0f;
#pragma unroll
    for (int j = 0; j < 16; ++j) {
        const float e = __expf(r[j] - rmax);
        r[j] = e;
        s += e;
    }
    s = wave_sum(s);
    if (lane == 0) sred[wav] = s;
    __syncthreads();
    if (tid == 0) {
        float t = 0.0f;
#pragma unroll
        for (int w = 0; w < NWAVE; ++w) t += sred[w];
        s_sum = t;
    }
    __syncthreads();

    const float inv     = 1.0f / s_sum;
    const int   acol    = gt[row];                         // scalar load
    const float anchorP = __expf(srow[acol] - rmax) * inv; // broadcast LDS read
    const float c       = MARGIN - anchorP;

    // ---- hinge sum: sum_j max(0, p_j - anchor + margin) ----
    float l = 0.0f;
#pragma unroll
    for (int j = 0; j < 16; ++j)
        l += fmaxf(fmaf(r[j], inv, c), 0.0f);

    l = wave_sum(l);
    if (lane == 0) sred[wav] = l;
    __syncthreads();
    if (tid == 0) {
        float t = 0.0f;
#pragma unroll
        for (int w = 0; w < NWAVE; ++w) t += sred[w];
        row_loss[row] = t;
    }
}

__global__ __launch_bounds__(BLK) void rankloss_reduce(
    const float* __restrict__ row_loss, float* __restrict__ out, int B)
{
    __shared__ float sred[NWAVE];
    float s = 0.0f;
    for (int i = threadIdx.x; i < B; i += BLK) s += row_loss[i];
    s = wave_sum(s);
    if ((threadIdx.x & 31) == 0) sred[threadIdx.x >> 5] = s;
    __syncthreads();
    if (threadIdx.x == 0) {
        float t = 0.0f;
#pragma unroll
        for (int w = 0; w < NWAVE; ++w) t += sred[w];
        out[0] = t / (float)B;
    }
}

extern "C" void kernel_launch(void* const* d_in, const int* in_sizes, int n_in,
                              void* d_out, int out_size, void* d_ws, size_t ws_size,
                              hipStream_t stream) {
    const float* X   = (const float*)d_in[0];  // [B, 4096] fp32
    const int*   gt  = (const int*)d_in[1];    // [B] indices
    float*       out = (float*)d_out;          // scalar
    float*       rl  = (float*)d_ws;           // [B] per-row sums (64 KB)

    const int B = in_sizes[1];                 // 16384

    rankloss_rows<<<dim3(B), dim3(BLK), 0, stream>>>(X, gt, rl);
    rankloss_reduce<<<dim3(1), dim3(BLK), 0, stream>>>(rl, out, B);
}